// LightGCN_41420664602963
// MI455X (gfx1250) — compile-verified
//
#include <hip/hip_runtime.h>
#include <hip/hip_bf16.h>
#include <stdint.h>

// LightGCN propagation on MI455X (gfx1250).
// Bandwidth-bound SpMM; whole working set (~132 MB) is L2-resident (192 MB).
// Edge tiles are staged to LDS with CDNA5 async global->LDS copies, double-
// buffered so tile N+1 streams in while tile N is processed.

#define TPB 256          // 8 wave32 per block
#define CHUNK 256        // edges per tile (3 KB of LDS: src, dst, val)
#define EDGE_LANES 16    // lanes cooperating per edge (64 floats / 4 per lane)

__global__ void lgcn_init_kernel(const float* __restrict__ user_emb,
                                 const float* __restrict__ item_emb,
                                 float* __restrict__ x_cur,
                                 float* __restrict__ out,
                                 int n_user_f, int total_f) {
    int i = blockIdx.x * blockDim.x + threadIdx.x;
    if (i >= total_f) return;
    float v = (i < n_user_f) ? user_emb[i] : item_emb[i - n_user_f];
    x_cur[i] = v;
    out[i]   = v;   // acc starts at the layer-0 embedding
}

__global__ void lgcn_zero_kernel(float* __restrict__ p, int n) {
    int i = blockIdx.x * blockDim.x + threadIdx.x;
    if (i < n) p[i] = 0.0f;
}

// d_out = (d_out + x_next) * scale   (scale = 1 inner layers, 0.25 at the end)
__global__ void lgcn_accum_kernel(float* __restrict__ out,
                                  const float* __restrict__ xn,
                                  int n, float scale) {
    int i = blockIdx.x * blockDim.x + threadIdx.x;
    if (i >= n) return;
    out[i] = (out[i] + xn[i]) * scale;
}

// Issue 6 async b128 copies staging one 256-edge tile (1KB src,1KB dst,1KB val)
// into LDS at byte offset lbase. Must be executed by a full wave (32 lanes).
__device__ __forceinline__ void stage_tile_async(uint32_t lbase, int lane32,
                                                 const int* esrc, const int* edst,
                                                 const float* eval, int ebase) {
    uint32_t l16 = lbase + (uint32_t)lane32 * 16u;
    uint64_t g0 = (uint64_t)(uintptr_t)(esrc + ebase) + (uint32_t)lane32 * 16u;
    uint64_t g1 = (uint64_t)(uintptr_t)(edst + ebase) + (uint32_t)lane32 * 16u;
    uint64_t g2 = (uint64_t)(uintptr_t)(eval + ebase) + (uint32_t)lane32 * 16u;
    asm volatile("global_load_async_to_lds_b128 %0, %1, off"
                 :: "v"(l16),          "v"(g0)        : "memory");
    asm volatile("global_load_async_to_lds_b128 %0, %1, off"
                 :: "v"(l16 + 512u),   "v"(g0 + 512u) : "memory");
    asm volatile("global_load_async_to_lds_b128 %0, %1, off"
                 :: "v"(l16 + 1024u),  "v"(g1)        : "memory");
    asm volatile("global_load_async_to_lds_b128 %0, %1, off"
                 :: "v"(l16 + 1536u),  "v"(g1 + 512u) : "memory");
    asm volatile("global_load_async_to_lds_b128 %0, %1, off"
                 :: "v"(l16 + 2048u),  "v"(g2)        : "memory");
    asm volatile("global_load_async_to_lds_b128 %0, %1, off"
                 :: "v"(l16 + 2560u),  "v"(g2 + 512u) : "memory");
}

// Gather-scale-scatter for one staged tile: 16 lanes per edge, each lane owns
// a float4 of the 64-float row; 4 no-return global_atomic_add_f32 per lane.
__device__ __forceinline__ void process_tile(const char* tile, int m,
                                             const float4* __restrict__ xc4,
                                             float* __restrict__ x_next,
                                             int lane, int eslot) {
    const int*   s_s = (const int*)tile;
    const int*   s_d = (const int*)(tile + 1024);
    const float* s_v = (const float*)(tile + 2048);
    #pragma unroll 4
    for (int it = 0; it < CHUNK / (TPB / EDGE_LANES); ++it) {   // 16 passes
        int el = it * (TPB / EDGE_LANES) + eslot;
        if (el >= m) break;
        int   s = s_s[el];
        int   d = s_d[el];
        float w = s_v[el];
        float4 v = xc4[(size_t)s * 16 + lane];                  // 16B gather
        float* dp = x_next + (size_t)d * 64 + lane * 4;
        atomicAdd(dp + 0, w * v.x);
        atomicAdd(dp + 1, w * v.y);
        atomicAdd(dp + 2, w * v.z);
        atomicAdd(dp + 3, w * v.w);
    }
}

// Each block owns two consecutive tiles, double-buffered in 6 KB of LDS.
__global__ void lgcn_scatter_kernel(const float* __restrict__ x_cur,
                                    float* __restrict__ x_next,
                                    const int* __restrict__ edge_src,
                                    const int* __restrict__ edge_dst,
                                    const float* __restrict__ edge_val,
                                    int n_edges) {
    extern __shared__ char smem[];                 // 2 x 3 KB tile buffers
    const int tid   = threadIdx.x;
    const int base0 = blockIdx.x * (2 * CHUNK);
    const int base1 = base0 + CHUNK;
    int m0 = n_edges - base0; m0 = m0 < CHUNK ? m0 : CHUNK;     // >= 1
    int m1 = n_edges - base1; m1 = m1 < CHUNK ? m1 : CHUNK;     // may be <= 0

    const int lane  = tid & (EDGE_LANES - 1);
    const int eslot = tid >> 4;
    const float4* xc4 = (const float4*)x_cur;

    if (m1 == CHUNK) {
        // Fast path: both tiles full. Pipeline: issue all 12 async copies,
        // wait for the first 6 (in-order completion), process tile 0 while
        // tile 1 streams in, then drain and process tile 1.
        if (tid < 32) {
            stage_tile_async(0u,    tid, edge_src, edge_dst, edge_val, base0);
            stage_tile_async(3072u, tid, edge_src, edge_dst, edge_val, base1);
            asm volatile("s_wait_asynccnt 6" ::: "memory");
        }
        __syncthreads();
        process_tile(smem, CHUNK, xc4, x_next, lane, eslot);
        if (tid < 32) {
            asm volatile("s_wait_asynccnt 0" ::: "memory");
        }
        __syncthreads();
        process_tile(smem + 3072, CHUNK, xc4, x_next, lane, eslot);
    } else {
        // Ragged tail: guarded cooperative staging (no OOB async reads).
        int*   s_s = (int*)smem;
        int*   s_d = (int*)(smem + 1024);
        float* s_v = (float*)(smem + 2048);
        for (int i = tid; i < m0; i += TPB) {
            s_s[i] = edge_src[base0 + i];
            s_d[i] = edge_dst[base0 + i];
            s_v[i] = edge_val[base0 + i];
        }
        int*   t_s = (int*)(smem + 3072);
        int*   t_d = (int*)(smem + 4096);
        float* t_v = (float*)(smem + 5120);
        for (int i = tid; i < m1; i += TPB) {
            t_s[i] = edge_src[base1 + i];
            t_d[i] = edge_dst[base1 + i];
            t_v[i] = edge_val[base1 + i];
        }
        __syncthreads();
        process_tile(smem, m0, xc4, x_next, lane, eslot);
        if (m1 > 0)
            process_tile(smem + 3072, m1, xc4, x_next, lane, eslot);
    }
}

extern "C" void kernel_launch(void* const* d_in, const int* in_sizes, int n_in,
                              void* d_out, int out_size, void* d_ws, size_t ws_size,
                              hipStream_t stream) {
    const float* user_emb = (const float*)d_in[0];
    const float* item_emb = (const float*)d_in[1];
    const int*   edge_src = (const int*)d_in[2];
    const int*   edge_dst = (const int*)d_in[3];
    const float* edge_val = (const float*)d_in[4];

    const int n_user_f = in_sizes[0];            // N_USERS * 64
    const int n_item_f = in_sizes[1];            // N_ITEMS * 64
    const int total_f  = n_user_f + n_item_f;    // N * 64
    const int n_edges  = in_sizes[2];

    float* out  = (float*)d_out;
    float* bufA = (float*)d_ws;                  // ping
    float* bufB = bufA + total_f;                // pong

    const int gridf  = (total_f + TPB - 1) / TPB;
    const int nblks  = (n_edges + 2 * CHUNK - 1) / (2 * CHUNK);

    lgcn_init_kernel<<<gridf, TPB, 0, stream>>>(user_emb, item_emb, bufA, out,
                                                n_user_f, total_f);

    float* xc = bufA;
    float* xn = bufB;
    for (int layer = 0; layer < 3; ++layer) {
        lgcn_zero_kernel<<<gridf, TPB, 0, stream>>>(xn, total_f);
        lgcn_scatter_kernel<<<nblks, TPB, 2 * 3072, stream>>>(
            xc, xn, edge_src, edge_dst, edge_val, n_edges);
        const float scale = (layer == 2) ? 0.25f : 1.0f;
        lgcn_accum_kernel<<<gridf, TPB, 0, stream>>>(out, xn, total_f, scale);
        float* t = xc; xc = xn; xn = t;
    }
}